// GumbelTopK_29480655520108
// MI455X (gfx1250) — compile-verified
//
#include <hip/hip_runtime.h>
#include <stdint.h>

// Problem constants from the reference
#define TK_N        8192      // n (row length)
#define TK_K        512       // top-k
#define TK_S        100       // monte-carlo samples
#define TK_THREADS  256       // 8 wave32s per block
#define TK_EPT      (TK_N / TK_THREADS)   // 32 elements per thread
#define TK_SPC      10        // samples per block
#define TK_CHUNKS   (TK_S / TK_SPC)       // 10 blocks per batch row
#define TK_KEY_SLOTS (TK_N + (TK_N >> 6)) // skewed key array (bank-conflict free)
#define TK_HCOPIES  4         // privatized histogram copies (2 waves each)
#define TK_HSTRIDE  257       // 256 bins + 1 skew word => copies on distinct banks

// Skewed LDS slot: conflict-free for both interleaved (e*256+t) and
// contiguous (t*32+i) access patterns on 64 banks.
__device__ __forceinline__ int kslot(int i) { return i + (i >> 6); }

// Order-preserving float -> u32 mapping (larger float => larger uint).
__device__ __forceinline__ uint32_t order_key(float v) {
  uint32_t u = __float_as_uint(v);
  return (u & 0x80000000u) ? ~u : (u | 0x80000000u);
}

// Issue one 32KB row load (8 x B128 per thread) global -> LDS via the CDNA5
// async data path. Each lane moves 16B per instruction; tracked on ASYNCcnt.
__device__ __forceinline__ void async_row_load(const float* __restrict__ gsrc,
                                               float* lds_dst, int tid) {
#pragma unroll
  for (int it = 0; it < 8; ++it) {
    const int elem = it * (TK_THREADS * 4) + tid * 4;
    const uint32_t laddr = (uint32_t)(uintptr_t)(lds_dst + elem);
    const float* g = gsrc + elem;
    asm volatile("global_load_async_to_lds_b128 %0, %1, off"
                 :: "v"(laddr), "v"(g)
                 : "memory");
  }
}

__global__ __launch_bounds__(TK_THREADS)
void gumbel_topk_count(const float* __restrict__ logits,
                       const float* __restrict__ uniform,
                       float* __restrict__ out) {
  extern __shared__ unsigned char smem_raw[];
  float*    ubuf0 = (float*)smem_raw;                  // 32 KB
  float*    ubuf1 = ubuf0 + TK_N;                      // 32 KB
  uint32_t* keys  = (uint32_t*)(ubuf1 + TK_N);         // ~32.5 KB (skewed)
  uint32_t* hist4 = keys + TK_KEY_SLOTS;               // 4 x 257 (privatized)
  uint32_t* hsum  = hist4 + TK_HCOPIES * TK_HSTRIDE;   // 256 combined / scan buf
  uint32_t* bc    = hsum + TK_THREADS;                 // broadcast scalars

  const int tid   = threadIdx.x;
  const int hcopy = tid >> 6;                          // 2 waves per copy
  const int b     = blockIdx.x / TK_CHUNKS;
  const int chunk = blockIdx.x % TK_CHUNKS;
  const int s0    = chunk * TK_SPC;

  // Per-thread logits (interleaved ownership: index e*256+tid). L2-resident.
  float lg[TK_EPT];
#pragma unroll
  for (int e = 0; e < TK_EPT; ++e)
    lg[e] = logits[(size_t)b * TK_N + e * TK_THREADS + tid];

  // Prime the async pipeline with sample s0.
  async_row_load(uniform + (size_t)(b * TK_S + s0) * TK_N, ubuf0, tid);

  for (int j = 0; j < TK_SPC; ++j) {
    float* cur = (j & 1) ? ubuf1 : ubuf0;
    float* nxt = (j & 1) ? ubuf0 : ubuf1;

    if (j + 1 < TK_SPC) {
      // Overlap: fetch sample j+1 while we process sample j.
      async_row_load(uniform + (size_t)(b * TK_S + s0 + j + 1) * TK_N, nxt, tid);
      asm volatile("s_wait_asynccnt 8" ::: "memory"); // first 8 (cur) done
    } else {
      asm volatile("s_wait_asynccnt 0" ::: "memory");
    }
    __syncthreads();  // cur buffer complete block-wide

    // ---- Gumbel perturbation + orderable keys into LDS ----
#pragma unroll
    for (int e = 0; e < TK_EPT; ++e) {
      const int i = e * TK_THREADS + tid;
      const float u = cur[i];
      const float g = -__logf(-__logf(u + 1e-20f) + 1e-20f);
      keys[kslot(i)] = order_key(lg[e] + g);   // TAU == 1.0
    }
    if (tid == 0) bc[1] = TK_K;   // k-remaining
    __syncthreads();

    // ---- MSB-first 8-bit radix select: exact K-th largest key ----
    uint32_t prefix = 0, pmask = 0;
#pragma unroll
    for (int shift = 24; shift >= 0; shift -= 8) {
      // zero all privatized copies
#pragma unroll
      for (int z = tid; z < TK_HCOPIES * TK_HSTRIDE; z += TK_THREADS)
        hist4[z] = 0;
      __syncthreads();
#pragma unroll
      for (int e = 0; e < TK_EPT; ++e) {
        const uint32_t kk = keys[kslot(e * TK_THREADS + tid)];
        if ((kk & pmask) == prefix)
          atomicAdd(&hist4[hcopy * TK_HSTRIDE + ((kk >> shift) & 255u)], 1u);
      }
      __syncthreads();
      // combine copies -> hsum[bin]
      {
        uint32_t c = 0;
#pragma unroll
        for (int w = 0; w < TK_HCOPIES; ++w) c += hist4[w * TK_HSTRIDE + tid];
        hsum[tid] = c;
      }
      __syncthreads();
      if (tid == 0) {
        uint32_t kRem = bc[1], cum = 0;
        int d = 255;
        for (; d > 0; --d) {
          const uint32_t c = hsum[d];
          if (cum + c >= kRem) break;
          cum += c;
        }
        bc[0] = (uint32_t)d;
        bc[1] = kRem - cum;
      }
      __syncthreads();
      prefix |= bc[0] << shift;
      pmask  |= 255u << shift;
      __syncthreads();
    }
    const uint32_t thr          = prefix;  // exact K-th largest key
    const uint32_t ties_allowed = bc[1];   // # threshold-valued keys in top-K

    // ---- Deterministic lowest-index-first tie ranking ----
    // Contiguous ownership for index order: thread t owns [t*32, t*32+32).
    uint32_t my_ties = 0;
#pragma unroll
    for (int i = 0; i < TK_EPT; ++i)
      if (keys[kslot(tid * TK_EPT + i)] == thr) my_ties++;
    hsum[tid] = my_ties;
    __syncthreads();
    for (int off = 1; off < TK_THREADS; off <<= 1) {  // inclusive scan
      const uint32_t v    = hsum[tid];
      const uint32_t addv = (tid >= off) ? hsum[tid - off] : 0u;
      __syncthreads();
      hsum[tid] = v + addv;
      __syncthreads();
    }
    uint32_t rank = hsum[tid] - my_ties;   // exclusive prefix of ties

    // ---- Emit: +1 for each selected index (exact small-int float adds) ----
    const size_t obase = (size_t)b * TK_N;
#pragma unroll
    for (int i = 0; i < TK_EPT; ++i) {
      const int idx = tid * TK_EPT + i;
      const uint32_t kk = keys[kslot(idx)];
      bool sel = kk > thr;
      if (kk == thr) { sel = (rank < ties_allowed); rank++; }
      if (sel) atomicAdd(&out[obase + idx], 1.0f);
    }
    __syncthreads();  // buffer `cur` may be re-filled next iteration
  }
}

__global__ void tk_zero(float* __restrict__ out, int n) {
  const int i = blockIdx.x * blockDim.x + threadIdx.x;
  if (i < n) out[i] = 0.0f;
}

__global__ void tk_scale(float* __restrict__ out, int n, float s) {
  const int i = blockIdx.x * blockDim.x + threadIdx.x;
  if (i < n) out[i] *= s;
}

extern "C" void kernel_launch(void* const* d_in, const int* in_sizes, int n_in,
                              void* d_out, int out_size, void* d_ws, size_t ws_size,
                              hipStream_t stream) {
  const float* logits  = (const float*)d_in[0];   // [B, 8192]
  const float* uniform = (const float*)d_in[1];   // [B, 100, 8192]
  float* out = (float*)d_out;                     // [B, 8192]

  const int B    = in_sizes[0] / TK_N;
  const int nOut = B * TK_N;

  tk_zero<<<(nOut + 255) / 256, 256, 0, stream>>>(out, nOut);

  const size_t smem = (size_t)(2 * TK_N) * sizeof(float)            // double buffer
                    + (size_t)TK_KEY_SLOTS * sizeof(uint32_t)       // skewed keys
                    + (size_t)(TK_HCOPIES * TK_HSTRIDE) * sizeof(uint32_t) // hist4
                    + (size_t)(TK_THREADS + 8) * sizeof(uint32_t);  // hsum + bc
  gumbel_topk_count<<<B * TK_CHUNKS, TK_THREADS, smem, stream>>>(logits, uniform, out);

  tk_scale<<<(nOut + 255) / 256, 256, 0, stream>>>(out, nOut, 1.0f / TK_S);
}